// CAM_Module_11020886082197
// MI455X (gfx1250) — compile-verified
//
#include <hip/hip_runtime.h>

// CAM module: energy = Q Q^T (K=65536), softmax(max-e), out = A Q, y = g*out + x
// B=2, C=128, N = 16*64*64 = 65536, all f32 (K=65536 accumulation => keep f32 WMMA).

typedef float v2f __attribute__((ext_vector_type(2)));
typedef float v8f __attribute__((ext_vector_type(8)));
typedef int   v2i_b64 __attribute__((vector_size(8)));   // b64 async payload type

#define BB 2
#define CC 128
#define NN 65536

// ---- CDNA5 async-to-LDS (ASYNCcnt) with graceful fallback --------------------
#if __has_builtin(__builtin_amdgcn_global_load_async_to_lds_b64) && \
    __has_builtin(__builtin_amdgcn_s_wait_asynccnt)
#define HAS_ASYNC 1
#define ASYNC_COPY_B64(g, l)                                                   \
    __builtin_amdgcn_global_load_async_to_lds_b64(                             \
        (__attribute__((address_space(1))) v2i_b64*)(g),                       \
        (__attribute__((address_space(3))) v2i_b64*)(l), 0, 0)
#define ASYNC_WAIT() __builtin_amdgcn_s_wait_asynccnt(0)
#else
#define HAS_ASYNC 0
#define ASYNC_COPY_B64(g, l) (*(v2f*)(l) = *(const v2f*)(g))
#define ASYNC_WAIT()
#endif

// ---------------- zero the energy scratch (2*128*128 floats) ----------------
__global__ void k_zero(float* __restrict__ e) {
    e[blockIdx.x * 256 + threadIdx.x] = 0.0f;
}

// ---------------- energy = Q Q^T, LDS-staged, WMMA f32 16x16x4 --------------
// grid: 2(b) * 16(32x32 tile) * 32(ksplit) = 1024 blocks, 256 threads (8 waves).
// Per block: double-buffered LDS stages of 64 K-columns for a 32(A)+32(B) row
// tile (async copies tracked by ASYNCcnt). 8 waves = 4 output 16x16 sub-tiles
// x 2-way K-split; 8 chained WMMAs per wave per stage; pair-reduce in LDS,
// then one global f32 atomic per C element.
#define SP 68   // LDS row stride (floats): bank step 4/row => conflict-free b64 frags
__global__ void k_energy(const float* __restrict__ x, float* __restrict__ energy) {
    const int bx   = blockIdx.x;
    const int ks   = bx & 31;
    const int tij  = (bx >> 5) & 15;
    const int b    = bx >> 9;
    const int ti   = tij >> 2;              // 32-row A tile index
    const int tj   = tij & 3;               // 32-row B tile index
    const int wave = threadIdx.x >> 5;
    const int lane = threadIdx.x & 31;
    const int l16  = lane & 15;
    const int koff = (lane >> 4) << 1;      // lanes 0-15: K0,K1 ; lanes 16-31: K2,K3

    const int tile = wave >> 1;             // 0..3 : which 16x16 sub-tile
    const int tm   = tile >> 1;             // sub-tile row (0/1)
    const int tn   = tile & 1;              // sub-tile col (0/1)
    const int ksub = wave & 1;              // K-half within a 64-col stage

    __shared__ float sA[2][32 * SP];
    __shared__ float sB[2][32 * SP];
    __shared__ float sred[4 * 256];

    const float* q  = x + (size_t)b * CC * NN;
    const float* qa = q + (size_t)(ti * 32) * NN;
    const float* qb = q + (size_t)(tj * 32) * NN;
    const int kb0 = ks * 2048;              // this block's K range: 2048 cols, 32 stages

    const int t    = threadIdx.x;
    // copy job: 1024 b64 chunks per matrix per stage -> 4 A + 4 B per thread
    const int crow[4] = { (t + 0)   >> 5, (t + 256) >> 5, (t + 512) >> 5, (t + 768) >> 5 };
    const int cseg[4] = { (t + 0)   & 31, (t + 256) & 31, (t + 512) & 31, (t + 768) & 31 };

    auto issue = [&](int stage, int buf) {
        const size_t kpos = (size_t)kb0 + (size_t)stage * 64;
#pragma unroll
        for (int c = 0; c < 4; ++c) {
            const int row = crow[c], seg = cseg[c];
            const size_t go = (size_t)row * NN + kpos + (size_t)seg * 2;
            const int    lo = row * SP + seg * 2;
            ASYNC_COPY_B64(qa + go, &sA[buf][lo]);
            ASYNC_COPY_B64(qb + go, &sB[buf][lo]);
        }
    };

    v8f acc = {};
    issue(0, 0);
    for (int s = 0; s < 32; ++s) {
        const int buf = s & 1;
        ASYNC_WAIT();                        // my async copies for stage s done
        __syncthreads();                     // everyone's copies visible
        if (s + 1 < 32) issue(s + 1, buf ^ 1);

        const float* aB = &sA[buf][(tm * 16 + l16) * SP + ksub * 32 + koff];
        const float* bB = &sB[buf][(tn * 16 + l16) * SP + ksub * 32 + koff];
#pragma unroll
        for (int k = 0; k < 32; k += 4) {
            v2f a  = *(const v2f*)(aB + k);
            v2f bb = *(const v2f*)(bB + k);
            acc = __builtin_amdgcn_wmma_f32_16x16x4_f32(
                      false, a, false, bb, (short)0, acc, false, false);
        }
        __syncthreads();                     // buf free before stage s+2 overwrites it
    }

    // pair-reduce the 2-way K-split, then atomic-accumulate
    if (ksub == 1) {
#pragma unroll
        for (int r = 0; r < 8; ++r) sred[tile * 256 + r * 32 + lane] = acc[r];
    }
    __syncthreads();
    if (ksub == 0) {
#pragma unroll
        for (int r = 0; r < 8; ++r) {
            const float s = acc[r] + sred[tile * 256 + r * 32 + lane];
            const int m = ti * 32 + tm * 16 + r + ((lane >> 4) << 3);
            const int n = tj * 32 + tn * 16 + (lane & 15);
            atomicAdd(&energy[((size_t)b * CC + m) * CC + n], s);
        }
    }
}

// ---------------- softmax(max(e) - e) per row, in place ---------------------
__global__ void k_softmax(float* __restrict__ e) {
    const int row  = blockIdx.x * 8 + (threadIdx.x >> 5);
    const int lane = threadIdx.x & 31;
    float* er = e + (size_t)row * CC;

    float v[4];
#pragma unroll
    for (int t = 0; t < 4; ++t) v[t] = er[lane + 32 * t];

    float mx = fmaxf(fmaxf(v[0], v[1]), fmaxf(v[2], v[3]));
#pragma unroll
    for (int o = 16; o > 0; o >>= 1) mx = fmaxf(mx, __shfl_xor(mx, o, 32));

    float en[4];
#pragma unroll
    for (int t = 0; t < 4; ++t) en[t] = mx - v[t];

    float mx2 = fmaxf(fmaxf(en[0], en[1]), fmaxf(en[2], en[3]));
#pragma unroll
    for (int o = 16; o > 0; o >>= 1) mx2 = fmaxf(mx2, __shfl_xor(mx2, o, 32));

    float p[4];
    float sum = 0.0f;
#pragma unroll
    for (int t = 0; t < 4; ++t) { p[t] = __expf(en[t] - mx2); sum += p[t]; }
#pragma unroll
    for (int o = 16; o > 0; o >>= 1) sum += __shfl_xor(sum, o, 32);

    const float inv = 1.0f / sum;
#pragma unroll
    for (int t = 0; t < 4; ++t) er[lane + 32 * t] = p[t] * inv;
}

// ---------------- out = gamma * (A @ Q) + x, WMMA f32 16x16x4 ---------------
// GEMM M=128, K=128, N=65536. grid: 2(b)*1024(64-wide N slice) blocks.
// wave = m-tile, so all 8 waves consume the same q cachelines back-to-back:
// q is streamed once per batch from L2. Prefetch B rows 8 K-steps ahead.
__global__ void k_av(const float* __restrict__ x, const float* __restrict__ attn,
                     const float* __restrict__ gamma, float* __restrict__ out) {
    const int bx   = blockIdx.x;
    const int nblk = bx & 1023;
    const int b    = bx >> 10;
    const int mt   = threadIdx.x >> 5;      // wave = M tile
    const int lane = threadIdx.x & 31;
    const int l16  = lane & 15;
    const int half = lane >> 4;
    const int koff = half << 1;

    const float* q    = x + (size_t)b * CC * NN;
    const float* arow = attn + ((size_t)b * CC + mt * 16 + l16) * CC;
    const int n0 = nblk * 64;

    v8f acc[4] = {v8f{}, v8f{}, v8f{}, v8f{}};

    for (int k = 0; k < CC; k += 4) {
        v2f a = *(const v2f*)(arow + k + koff);            // contiguous K in attn row
        const float* bbase = q + (size_t)(k + koff) * NN + n0 + l16;
        if (k < 120) __builtin_prefetch(bbase + (size_t)8 * NN, 0, 1);
#pragma unroll
        for (int t = 0; t < 4; ++t) {
            v2f bb;
            bb.x = bbase[t * 16];                          // B[k+koff][n]
            bb.y = bbase[(size_t)NN + t * 16];             // B[k+koff+1][n]
            acc[t] = __builtin_amdgcn_wmma_f32_16x16x4_f32(
                         false, a, false, bb, (short)0, acc[t], false, false);
        }
    }

    const float g = gamma[0];
#pragma unroll
    for (int t = 0; t < 4; ++t) {
#pragma unroll
        for (int r = 0; r < 8; ++r) {
            const int m = mt * 16 + r + (half << 3);
            const size_t idx = ((size_t)b * CC + m) * NN + n0 + t * 16 + l16;
            out[idx] = g * acc[t][r] + x[idx];
        }
    }
}

// ---------------- launch ----------------------------------------------------
extern "C" void kernel_launch(void* const* d_in, const int* in_sizes, int n_in,
                              void* d_out, int out_size, void* d_ws, size_t ws_size,
                              hipStream_t stream) {
    const float* x     = (const float*)d_in[0];
    const float* gamma = (const float*)d_in[1];
    float* out    = (float*)d_out;
    float* energy = (float*)d_ws;            // 2*128*128 f32 = 128 KB scratch

    k_zero   <<<128,  256, 0, stream>>>(energy);
    k_energy <<<1024, 256, 0, stream>>>(x, energy);
    k_softmax<<<32,   256, 0, stream>>>(energy);
    k_av     <<<2048, 256, 0, stream>>>(x, energy, gamma, out);
}